// ListMLELoss_65377992179980
// MI455X (gfx1250) — compile-verified
//
#include <hip/hip_runtime.h>

#define LROW 500
#define WPB  8            // waves per block

#define AS3 __attribute__((address_space(3)))
#define AS1 __attribute__((address_space(1)))

typedef __attribute__((__vector_size__(4 * sizeof(int)))) int v4i;

#define LOG2E 1.44269504088896340736f
#define LN2   0.69314718055994530942f

__device__ __forceinline__ float fast_exp2(float x) {
#if defined(__has_builtin) && __has_builtin(__builtin_amdgcn_exp2f)
  return __builtin_amdgcn_exp2f(x);
#else
  return exp2f(x);
#endif
}
__device__ __forceinline__ float fast_log2(float x) {
#if defined(__has_builtin) && __has_builtin(__builtin_amdgcn_logf)
  return __builtin_amdgcn_logf(x);   // v_log_f32 = log2
#else
  return log2f(x);
#endif
}

__device__ __forceinline__ float wred_add(float v) {
#pragma unroll
  for (int d = 16; d >= 1; d >>= 1) v += __shfl_xor(v, d, 32);
  return v;
}
__device__ __forceinline__ float wred_max(float v) {
#pragma unroll
  for (int d = 16; d >= 1; d >>= 1) v = fmaxf(v, __shfl_xor(v, d, 32));
  return v;
}

__global__ __launch_bounds__(WPB * 32)
void listmle_rows(const float* __restrict__ scores,
                  const int*   __restrict__ ranking,
                  float*       __restrict__ rowLoss,
                  int nrows)
{
  __shared__ float sBuf[WPB][512];   // 2KB staging per wave
  const int lane = threadIdx.x & 31;
  const int w    = threadIdx.x >> 5;
  const int row  = blockIdx.x * WPB + w;
  if (row >= nrows) return;

  const float* rowS = scores  + (size_t)row * LROW;
  const int*   rowR = ranking + (size_t)row * LROW;

  // ---- async stage: scores row -> LDS, 4x B128 per wave (512B per op) ----
#pragma unroll
  for (int c = 0; c < 4; ++c) {
    int off = c * 512 + lane * 16;                    // byte offset in row
    if (off > (LROW * 4 - 16)) off = LROW * 4 - 16;   // clamp tail inside row
    const char* gp = (const char*)rowS + off;
    unsigned int lp =
        (unsigned int)(unsigned long long)(const void*)((const char*)&sBuf[w][0] + off);
#if defined(__has_builtin) && __has_builtin(__builtin_amdgcn_global_load_async_to_lds_b128)
    __builtin_amdgcn_global_load_async_to_lds_b128(
        (AS1 v4i*)(unsigned long long)gp, (AS3 v4i*)lp, 0, 0);
#else
    asm volatile("global_load_async_to_lds_b128 %0, %1, off"
                 :: "v"(lp), "v"(gp) : "memory");
#endif
  }

  // ---- ranking loads overlap the copy: lane owns elements [16*lane, 16*lane+16) ----
  // Tail clamp duplicates real elements 496..499 -> gathered g stays a genuine
  // row value, so the row-max needs no masking.
  int rkv[16];
#pragma unroll
  for (int q = 0; q < 4; ++q) {
    int i4 = lane * 4 + q;            // int4 index within row (125 valid int4s)
    if (i4 > 124) i4 = 124;
    int4 r = ((const int4*)rowR)[i4];
    rkv[4 * q + 0] = r.x; rkv[4 * q + 1] = r.y;
    rkv[4 * q + 2] = r.z; rkv[4 * q + 3] = r.w;
  }

#if defined(__has_builtin) && __has_builtin(__builtin_amdgcn_s_wait_asynccnt)
  __builtin_amdgcn_s_wait_asynccnt(0);
#else
  asm volatile("s_wait_asynccnt 0" ::: "memory");
  asm volatile("" ::: "memory");
#endif

  // ---- gather g from LDS; unguarded max, guarded sum(g[0..L-2]) ----
  float gv[16];
  float mx = -3.402823466e38f;
  float sg = 0.f;
#pragma unroll
  for (int k = 0; k < 16; ++k) {
    int idx = lane * 16 + k;
    float g = sBuf[w][rkv[k]];        // always in [0,500): safe, real row value
    gv[k] = g;
    mx = fmaxf(mx, g);                // duplicates can't exceed the true max
    if (idx < LROW - 1) sg += g;      // folded unconditional for k <= 2
  }
  mx = wred_max(mx);
  const float nmx2 = -mx * LOG2E;

  // ---- e = 2^(g*log2e - mx*log2e) (fma + exp2), zero invalid tail ----
#pragma unroll
  for (int k = 0; k < 16; ++k) {
    int idx = lane * 16 + k;
    float e = fast_exp2(__fmaf_rn(gv[k], LOG2E, nmx2));
    gv[k] = (idx < LROW) ? e : 0.f;   // folded unconditional for k <= 3
  }
#pragma unroll
  for (int k = 14; k >= 0; --k) gv[k] += gv[k + 1];   // segment suffix sums

  // ---- one exclusive suffix scan of the 32 lane totals ----
  float after = __shfl_down(gv[0], 1, 32);
  if (lane == 31) after = 0.f;
#pragma unroll
  for (int d = 1; d < 32; d <<= 1) {
    float t = __shfl_down(after, d, 32);
    if (lane + d < 32) after += t;
  }
  // after[lane] = sum of totals of lanes > lane

  // ---- accumulate log2(T); scale by ln2 once at the end ----
  float lsum2 = 0.f;
#pragma unroll
  for (int k = 0; k < 16; ++k) {
    int idx = lane * 16 + k;
    if (idx < LROW - 1) lsum2 += fast_log2(gv[k] + after);
  }
  float red = wred_add(__fmaf_rn(lsum2, LN2, -sg));

  if (lane == 0)
    rowLoss[row] = (float)(LROW - 1) * mx + red;
}

__global__ void reduce_mean_kernel(const float* __restrict__ v,
                                   float* __restrict__ out,
                                   int n, float scale)
{
  __shared__ float red[32];
  int t = threadIdx.x;
  float acc = 0.f;
  for (int i = t; i < n; i += 1024) acc += v[i];
  acc = wred_add(acc);
  if ((t & 31) == 0) red[t >> 5] = acc;
  __syncthreads();
  if (t < 32) {
    float a = red[t];
    a = wred_add(a);
    if (t == 0) out[0] = a * scale;
  }
}

extern "C" void kernel_launch(void* const* d_in, const int* in_sizes, int n_in,
                              void* d_out, int out_size, void* d_ws, size_t ws_size,
                              hipStream_t stream) {
  (void)n_in; (void)out_size; (void)ws_size;
  const float* scores  = (const float*)d_in[0];
  const int*   ranking = (const int*)d_in[1];
  const int    B       = in_sizes[0] / LROW;

  float* rowLoss = (float*)d_ws;                 // B floats (256 KB)
  int blocks = (B + WPB - 1) / WPB;
  listmle_rows<<<blocks, WPB * 32, 0, stream>>>(scores, ranking, rowLoss, B);
  reduce_mean_kernel<<<1, 1024, 0, stream>>>(rowLoss, (float*)d_out, B, 1.0f / (float)B);
}